// LightGCN_16733192585495
// MI455X (gfx1250) — compile-verified
//
#include <hip/hip_runtime.h>

#define EMB_DIM 64
#define THREADS 256
#define EPB 256 // edges staged per block in the propagation kernel

#if __has_builtin(__builtin_amdgcn_global_load_async_to_lds_b32) && \
    __has_builtin(__builtin_amdgcn_s_wait_asynccnt)
#define HAVE_ASYNC_LDS 1
#else
#define HAVE_ASYNC_LDS 0
#endif

// ---------------- utility kernels ----------------

__global__ __launch_bounds__(THREADS) void k_zero_f4(float4* __restrict__ p, long n4) {
  long i = (long)blockIdx.x * blockDim.x + threadIdx.x;
  if (i < n4) p[i] = make_float4(0.f, 0.f, 0.f, 0.f);
}

// deg[col[e]] += 1
__global__ __launch_bounds__(THREADS) void k_deg(const int* __restrict__ col,
                                                 float* __restrict__ deg, int E) {
  int e = blockIdx.x * blockDim.x + threadIdx.x;
  if (e < E) atomicAdd(&deg[col[e]], 1.0f);
}

// deg -> deg^{-1/2} in place (0 stays 0)
__global__ __launch_bounds__(THREADS) void k_dis(float* __restrict__ deg, int N) {
  int n = blockIdx.x * blockDim.x + threadIdx.x;
  if (n < N) {
    float d = deg[n];
    deg[n] = (d > 0.f) ? rsqrtf(d) : 0.f;
  }
}

// norm[e] = dis[row[e]] * dis[col[e]]
__global__ __launch_bounds__(THREADS) void k_norm(const int* __restrict__ row,
                                                  const int* __restrict__ col,
                                                  const float* __restrict__ dis,
                                                  float* __restrict__ norm, int E) {
  int e = blockIdx.x * blockDim.x + threadIdx.x;
  if (e < E) norm[e] = dis[row[e]] * dis[col[e]];
}

// acc = emb  (float4)
__global__ __launch_bounds__(THREADS) void k_copy_f4(const float4* __restrict__ src,
                                                     float4* __restrict__ dst, long n4) {
  long i = (long)blockIdx.x * blockDim.x + threadIdx.x;
  if (i < n4) dst[i] = src[i];
}

// acc += src  (float4)
__global__ __launch_bounds__(THREADS) void k_acc_f4(const float4* __restrict__ src,
                                                    float4* __restrict__ acc, long n4) {
  long i = (long)blockIdx.x * blockDim.x + threadIdx.x;
  if (i < n4) {
    float4 a = acc[i], s = src[i];
    a.x += s.x; a.y += s.y; a.z += s.z; a.w += s.w;
    acc[i] = a;
  }
}

// acc += src; z = 0   (fused: prepare next scatter target while accumulating)
__global__ __launch_bounds__(THREADS) void k_acc_zero_f4(const float4* __restrict__ src,
                                                         float4* __restrict__ acc,
                                                         float4* __restrict__ z, long n4) {
  long i = (long)blockIdx.x * blockDim.x + threadIdx.x;
  if (i < n4) {
    float4 a = acc[i], s = src[i];
    a.x += s.x; a.y += s.y; a.z += s.z; a.w += s.w;
    acc[i] = a;
    z[i] = make_float4(0.f, 0.f, 0.f, 0.f);
  }
}

// acc = (acc + src) * 0.25  (final layer-mean over x^0..x^3)
__global__ __launch_bounds__(THREADS) void k_final_f4(const float4* __restrict__ src,
                                                      float4* __restrict__ acc, long n4) {
  long i = (long)blockIdx.x * blockDim.x + threadIdx.x;
  if (i < n4) {
    float4 a = acc[i], s = src[i];
    a.x = (a.x + s.x) * 0.25f; a.y = (a.y + s.y) * 0.25f;
    a.z = (a.z + s.z) * 0.25f; a.w = (a.w + s.w) * 0.25f;
    acc[i] = a;
  }
}

// ---------------- propagation: xout[col] += norm * xin[row] ----------------
// Block stages EPB edge records {row, col, norm} into LDS via the CDNA5 async
// global->LDS path (interleaved int4 records -> one ds_load_b128 per edge on
// the consumer side, conflict-free 16-lane broadcast). Then 16 lanes cooperate
// per edge: each lane gathers a float4 (256B coalesced per edge row) and
// scatter-adds with 4 global f32 atomics (L2-resident destination).
__global__ __launch_bounds__(THREADS) void k_prop(const int* __restrict__ row,
                                                  const int* __restrict__ col,
                                                  const float* __restrict__ norm,
                                                  const float* __restrict__ xin,
                                                  float* __restrict__ xout, int E) {
  __shared__ int4 s_edge[EPB]; // .x=row  .y=col  .z=norm bits  .w=pad

  const int t    = threadIdx.x;
  const int base = blockIdx.x * EPB;
  const int gE   = base + t;

#if HAVE_ASYNC_LDS
  if (gE < E) {
    int* sp = reinterpret_cast<int*>(&s_edge[t]);
    __builtin_amdgcn_global_load_async_to_lds_b32((int*)(row + gE), sp + 0, 0, 0);
    __builtin_amdgcn_global_load_async_to_lds_b32((int*)(col + gE), sp + 1, 0, 0);
    __builtin_amdgcn_global_load_async_to_lds_b32((int*)(norm + gE), sp + 2, 0, 0);
  }
  __builtin_amdgcn_s_wait_asynccnt(0);
  __syncthreads();
#else
  if (gE < E) {
    int* sp = reinterpret_cast<int*>(&s_edge[t]);
    sp[0] = row[gE];
    sp[1] = col[gE];
    sp[2] = __float_as_int(norm[gE]);
  }
  __syncthreads();
#endif

  const int q  = (t & 15) * 4; // dim quad this lane owns
  const int eg = t >> 4;       // edge sub-group 0..15

  if (base + EPB <= E) {
    // fast path: full tile, no bounds checks in the hot loop
#pragma unroll
    for (int k = 0; k < 16; ++k) {
      const int4  ed = s_edge[eg * 16 + k];
      const float n  = __int_as_float(ed.z);
      const float4 v = *reinterpret_cast<const float4*>(xin + (long)ed.x * EMB_DIM + q);
      float* dst = xout + (long)ed.y * EMB_DIM + q;
      atomicAdd(dst + 0, v.x * n);
      atomicAdd(dst + 1, v.y * n);
      atomicAdd(dst + 2, v.z * n);
      atomicAdd(dst + 3, v.w * n);
    }
  } else {
#pragma unroll
    for (int k = 0; k < 16; ++k) {
      const int eL = eg * 16 + k;
      if (base + eL < E) {
        const int4  ed = s_edge[eL];
        const float n  = __int_as_float(ed.z);
        const float4 v = *reinterpret_cast<const float4*>(xin + (long)ed.x * EMB_DIM + q);
        float* dst = xout + (long)ed.y * EMB_DIM + q;
        atomicAdd(dst + 0, v.x * n);
        atomicAdd(dst + 1, v.y * n);
        atomicAdd(dst + 2, v.z * n);
        atomicAdd(dst + 3, v.w * n);
      }
    }
  }
}

// ---------------- launch ----------------

extern "C" void kernel_launch(void* const* d_in, const int* in_sizes, int n_in,
                              void* d_out, int out_size, void* d_ws, size_t ws_size,
                              hipStream_t stream) {
  const int*   edge = (const int*)d_in[0];   // [2, E] flat
  const float* emb  = (const float*)d_in[1]; // [N, 64]
  const int E = in_sizes[0] / 2;
  const int N = in_sizes[1] / EMB_DIM;

  const int* rowp = edge;
  const int* colp = edge + E;

  char*  ws   = (char*)d_ws;
  const size_t xbytes = (size_t)N * EMB_DIM * sizeof(float);
  float* A    = (float*)(ws);
  float* B    = (float*)(ws + xbytes);
  float* deg  = (float*)(ws + 2 * xbytes);
  float* nrm  = (float*)(ws + 2 * xbytes + (size_t)N * sizeof(float));
  float* out  = (float*)d_out;

  const long nX4   = (long)N * EMB_DIM / 4;                 // float4 count of one x buffer
  const long zN4   = ((long)2 * N * EMB_DIM + N + 3) / 4;   // A + B + deg contiguous
  const int  gE    = (E + THREADS - 1) / THREADS;
  const int  gN    = (N + THREADS - 1) / THREADS;
  const int  gX4   = (int)((nX4 + THREADS - 1) / THREADS);
  const int  gZ4   = (int)((zN4 + THREADS - 1) / THREADS);
  const int  gProp = (E + EPB - 1) / EPB;

  // 1) zero A, B, deg (contiguous region at ws start)
  k_zero_f4<<<gZ4, THREADS, 0, stream>>>((float4*)ws, zN4);
  // 2) degree of col targets
  k_deg<<<gE, THREADS, 0, stream>>>(colp, deg, E);
  // 3) deg -> deg^{-1/2}
  k_dis<<<gN, THREADS, 0, stream>>>(deg, N);
  // 4) per-edge norm
  k_norm<<<gE, THREADS, 0, stream>>>(rowp, colp, deg, nrm, E);
  // 5) acc = x^0
  k_copy_f4<<<gX4, THREADS, 0, stream>>>((const float4*)emb, (float4*)out, nX4);

  // layer 1: emb -> A ; acc += A              (B already zeroed)
  k_prop<<<gProp, THREADS, 0, stream>>>(rowp, colp, nrm, emb, A, E);
  k_acc_f4<<<gX4, THREADS, 0, stream>>>((const float4*)A, (float4*)out, nX4);
  // layer 2: A -> B ; acc += B ; A = 0
  k_prop<<<gProp, THREADS, 0, stream>>>(rowp, colp, nrm, A, B, E);
  k_acc_zero_f4<<<gX4, THREADS, 0, stream>>>((const float4*)B, (float4*)out, (float4*)A, nX4);
  // layer 3: B -> A ; acc = (acc + A) / 4
  k_prop<<<gProp, THREADS, 0, stream>>>(rowp, colp, nrm, B, A, E);
  k_final_f4<<<gX4, THREADS, 0, stream>>>((const float4*)A, (float4*)out, nX4);
}